// LesionRegionSelector_26439818674305
// MI455X (gfx1250) — compile-verified
//
#include <hip/hip_runtime.h>
#include <math.h>

typedef __attribute__((ext_vector_type(2))) float v2f;
typedef __attribute__((ext_vector_type(4))) float f4;
typedef __attribute__((ext_vector_type(8))) float v8f;

#define EPS 1e-8f
#define NB 64
#define NP 4096
#define ND 512
#define NC 16
#define TOPK 16
#define TILE_M 16
#define LSTRIDE 516   // 512 + 4 pad: 16B-aligned rows, stride%64==4 -> conflict-free banks

// ---------------------------------------------------------------------------
// Kernel 1: pbar[b][d] = (1/16) * sum_c proto[b][c][d] / (||proto[b][c]|| + eps)
// ---------------------------------------------------------------------------
__global__ __launch_bounds__(256) void proto_mean_kernel(const float* __restrict__ protos,
                                                         float* __restrict__ pbar) {
    __shared__ float red[256];
    const int b = blockIdx.x;
    const int t = threadIdx.x;
    const float* pb = protos + (size_t)b * NC * ND;
    float accx = 0.f, accy = 0.f;
    for (int c = 0; c < NC; ++c) {
        float2 v = ((const float2*)(pb + c * ND))[t];   // 256 threads x float2 = 512 floats
        red[t] = v.x * v.x + v.y * v.y;
        __syncthreads();
        for (int s = 128; s > 0; s >>= 1) {
            if (t < s) red[t] += red[t + s];
            __syncthreads();
        }
        float rn = 1.0f / (sqrtf(red[0]) + EPS);
        __syncthreads();
        accx += v.x * rn;
        accy += v.y * rn;
    }
    float2 o;
    o.x = accx * (1.0f / 16.0f);
    o.y = accy * (1.0f / 16.0f);
    ((float2*)(pbar + (size_t)b * ND))[t] = o;
}

// ---------------------------------------------------------------------------
// Kernel 2: sim[b][p] = dot(lf[b][p], pbar[b]) / (||lf[b][p]|| + eps)
// One block = 16 patches. Dot via V_WMMA_F32_16X16X4_F32 (B = pbar broadcast).
// ---------------------------------------------------------------------------
__global__ __launch_bounds__(256) void sim_kernel(const float* __restrict__ lf,
                                                  const float* __restrict__ pbar_g,
                                                  float* __restrict__ sim) {
    __shared__ float tile[TILE_M * LSTRIDE];   // 33 KB staged patch tile
    __shared__ float pb[ND];
    __shared__ float sumsq[TILE_M];
    __shared__ float part[8 * TILE_M];

    const int t     = threadIdx.x;
    const int b     = blockIdx.y;
    const int tile0 = blockIdx.x * TILE_M;
    const int lane  = t & 31;
    const int wave  = t >> 5;

    if (t < TILE_M) sumsq[t] = 0.f;
    {   // pbar -> LDS
        float2 v = ((const float2*)(pbar_g + (size_t)b * ND))[t];
        pb[2 * t]     = v.x;
        pb[2 * t + 1] = v.y;
    }
    __syncthreads();

    // Stage 16 rows x 512 f32; 16 threads per row, float4-vectorized, coalesced.
    {
        const int row = t >> 4;     // 0..15
        const int l16 = t & 15;
        const float* src = lf + (((size_t)b * NP) + tile0 + row) * ND;
        float ss = 0.f;
        for (int j = 0; j < 8; ++j) {
            int c4 = l16 + 16 * j;              // float4 column 0..127
            f4 v = ((const f4*)src)[c4];
            ss += v.x * v.x + v.y * v.y + v.z * v.z + v.w * v.w;
            *(f4*)&tile[row * LSTRIDE + 4 * c4] = v;
        }
        atomicAdd(&sumsq[row], ss);             // ds_add_f32
    }
    __syncthreads();

    // WMMA: D(16x16) += A(16x4) * B(4x16); wave w covers K = w*64 .. w*64+63.
    // A: lane L<16 -> (M=L, K=kb..kb+1); lane>=16 -> (M=L-16, K=kb+2..kb+3).
    // B: pbar broadcast to all 16 columns (same per-lane K layout).
    v8f acc = {};
    const int arow = t & 15;
    const int koff = (lane < 16) ? 0 : 2;
    for (int kk = wave * 16; kk < wave * 16 + 16; ++kk) {
        const int kb = 4 * kk + koff;
        v2f a  = *(const v2f*)&tile[arow * LSTRIDE + kb];
        v2f bb = *(const v2f*)&pb[kb];
        acc = __builtin_amdgcn_wmma_f32_16x16x4_f32(
            /*neg_a=*/false, a, /*neg_b=*/false, bb,
            /*c_mod=*/(short)0, acc, /*reuse_a=*/false, /*reuse_b=*/false);
    }

    // Extract column N=0: lane0 holds M=r (VGPR r), lane16 holds M=r+8.
    if (lane == 0) {
        for (int r = 0; r < 8; ++r) part[wave * 16 + r] = acc[r];
    } else if (lane == 16) {
        for (int r = 0; r < 8; ++r) part[wave * 16 + 8 + r] = acc[r];
    }
    __syncthreads();

    if (t < TILE_M) {
        float dot = 0.f;
        for (int w = 0; w < 8; ++w) dot += part[w * 16 + t];
        sim[(size_t)b * NP + tile0 + t] = dot / (sqrtf(sumsq[t]) + EPS);
    }
}

// ---------------------------------------------------------------------------
// Kernel 3: per-batch top-16 / bottom-16 (tie -> lower index, matching
// jax.lax.top_k), then gather selected feature rows + write indices.
// ---------------------------------------------------------------------------
__global__ __launch_bounds__(256) void select_gather_kernel(const float* __restrict__ lf,
                                                            const float* __restrict__ sim,
                                                            float* __restrict__ out) {
    __shared__ float sv[NP];     // copy for top pass (masked with -MAX)
    __shared__ float sv2[NP];    // copy for bottom pass (masked with +MAX)
    __shared__ float rv[256];
    __shared__ int   ri[256];
    __shared__ int   selidx[32];

    const int b = blockIdx.x;
    const int t = threadIdx.x;
    const float* srow = sim + (size_t)b * NP;
    for (int j = 0; j < NP / 256; ++j) {
        float v = srow[t + 256 * j];
        sv[t + 256 * j]  = v;
        sv2[t + 256 * j] = v;
    }
    __syncthreads();

    const float NEGF = -3.402823466e38f, POSF = 3.402823466e38f;

    // top-k: iterative argmax with masking
    for (int k = 0; k < TOPK; ++k) {
        float bv = NEGF; int bi = 0x7fffffff;
        for (int j = 0; j < NP / 256; ++j) {
            int i = t + 256 * j;
            float v = sv[i];
            if (v > bv) { bv = v; bi = i; }     // ascending scan -> lowest index on tie
        }
        rv[t] = bv; ri[t] = bi;
        __syncthreads();
        for (int s = 128; s > 0; s >>= 1) {
            if (t < s) {
                float v2 = rv[t + s]; int i2 = ri[t + s];
                if (v2 > rv[t] || (v2 == rv[t] && i2 < ri[t])) { rv[t] = v2; ri[t] = i2; }
            }
            __syncthreads();
        }
        if (t == 0) { selidx[k] = ri[0]; sv[ri[0]] = NEGF; }
        __syncthreads();
    }

    // bottom-k: iterative argmin with masking
    for (int k = 0; k < TOPK; ++k) {
        float bv = POSF; int bi = 0x7fffffff;
        for (int j = 0; j < NP / 256; ++j) {
            int i = t + 256 * j;
            float v = sv2[i];
            if (v < bv) { bv = v; bi = i; }
        }
        rv[t] = bv; ri[t] = bi;
        __syncthreads();
        for (int s = 128; s > 0; s >>= 1) {
            if (t < s) {
                float v2 = rv[t + s]; int i2 = ri[t + s];
                if (v2 < rv[t] || (v2 == rv[t] && i2 < ri[t])) { rv[t] = v2; ri[t] = i2; }
            }
            __syncthreads();
        }
        if (t == 0) { selidx[16 + k] = ri[0]; sv2[ri[0]] = POSF; }
        __syncthreads();
    }

    // Gather 32 rows x 512 floats; 8 threads per row, float4-vectorized.
    const size_t F = (size_t)NB * TOPK * ND;    // 524288 (size of top_feat block)
    const int r  = t >> 3;                      // 0..31: 0-15 top, 16-31 bottom
    const int l8 = t & 7;
    const int idx = selidx[r];
    const f4* src = (const f4*)(lf + (((size_t)b * NP) + idx) * ND);
    float* dstp = (r < TOPK)
        ? (out + ((size_t)b * TOPK + r) * ND)
        : (out + F + ((size_t)b * TOPK + (r - TOPK)) * ND);
    f4* dst = (f4*)dstp;
    for (int j = 0; j < 16; ++j) {
        int c4 = l8 + 8 * j;
        dst[c4] = src[c4];
    }

    // indices (as float, reference return order: top_idx then bot_idx)
    if (t < TOPK) {
        out[2 * F + (size_t)b * TOPK + t] = (float)selidx[t];
    } else if (t < 2 * TOPK) {
        out[2 * F + (size_t)NB * TOPK + (size_t)b * TOPK + (t - TOPK)] = (float)selidx[t];
    }
}

// ---------------------------------------------------------------------------
extern "C" void kernel_launch(void* const* d_in, const int* in_sizes, int n_in,
                              void* d_out, int out_size, void* d_ws, size_t ws_size,
                              hipStream_t stream) {
    (void)in_sizes; (void)n_in; (void)out_size; (void)ws_size;
    const float* lf = (const float*)d_in[0];   // [64, 4096, 512] f32
    const float* pr = (const float*)d_in[1];   // [64, 16, 512]  f32
    float* out  = (float*)d_out;
    float* pbar = (float*)d_ws;                // 64*512 floats
    float* sim  = pbar + (size_t)NB * ND;      // 64*4096 floats

    proto_mean_kernel<<<dim3(NB), dim3(256), 0, stream>>>(pr, pbar);
    sim_kernel<<<dim3(NP / TILE_M, NB), dim3(256), 0, stream>>>(lf, pbar, sim);
    select_gather_kernel<<<dim3(NB), dim3(256), 0, stream>>>(lf, sim, out);
}